// HT2SPHERE_41875931136702
// MI455X (gfx1250) — compile-verified
//
#include <hip/hip_runtime.h>
#include <hip/hip_bf16.h>
#include <stdint.h>

// Problem constants (from reference)
constexpr int Bc  = 4;
constexpr int Cc  = 128;
constexpr int Hc  = 184;
constexpr int Wc  = 180;
constexpr int HWc = Hc * Wc;          // 33120
constexpr int Pc  = 32768;            // points
constexpr int Vc  = 16;               // votes per point

constexpr int PTS_PER_BLOCK = 64;     // 8 waves * 8 points
constexpr int THREADS       = 256;

typedef __attribute__((ext_vector_type(4))) unsigned int v4u;
typedef __attribute__((ext_vector_type(4))) int          v4i;
typedef __attribute__((ext_vector_type(8))) int          v8i;

// ---------------------------------------------------------------------------
// Kernel 1: flatten int64 (h,w) pairs -> int32 flat index (done once, reused
// by all 512 (b,c) planes). mapping is read-once -> non-temporal loads.
// ---------------------------------------------------------------------------
__global__ __launch_bounds__(256)
void ht2s_idx_kernel(const long long* __restrict__ mapping,
                     int* __restrict__ flat_idx, int n)
{
    int i = blockIdx.x * blockDim.x + threadIdx.x;
    if (i < n) {
        long long h = __builtin_nontemporal_load(&mapping[2 * i + 0]);
        long long w = __builtin_nontemporal_load(&mapping[2 * i + 1]);
        flat_idx[i] = (int)(h * (long long)Wc + w);
    }
}

// ---------------------------------------------------------------------------
// Kernel 2: transpose feats (B, C, HW) -> featsT (B, HW, C) so that a gather
// of one spatial index pulls 128 contiguous channel floats (512 B burst).
// Tile loads go global -> LDS directly via gfx1250 async-to-LDS (ASYNCcnt),
// bypassing the VGPR staging of a classic transpose.
// ---------------------------------------------------------------------------
__global__ __launch_bounds__(256)
void ht2s_transpose_kernel(const float* __restrict__ feats,
                           float* __restrict__ featsT)
{
    __shared__ float tile[32][33];
    const int b  = blockIdx.z;
    const int x0 = blockIdx.x * 32;   // HW tile origin (HW = 33120 = 1035*32)
    const int c0 = blockIdx.y * 32;   // channel tile origin (128 = 4*32)
    const int tx = threadIdx.x;       // 0..31
    const int ty = threadIdx.y;       // 0..7

    const float* src = feats + ((size_t)b * Cc + c0) * HWc + x0;
#pragma unroll
    for (int j = 0; j < 4; ++j) {
        const float* g = src + (size_t)(ty + 8 * j) * HWc + tx;
        unsigned lds_off = (unsigned)(size_t)(&tile[ty + 8 * j][tx]);
        asm volatile("global_load_async_to_lds_b32 %0, %1, off"
                     :: "v"(lds_off), "v"(g) : "memory");
    }
    asm volatile("s_wait_asynccnt 0" ::: "memory");
    __syncthreads();

    float* dst = featsT + ((size_t)b * HWc + x0) * Cc + c0;
#pragma unroll
    for (int j = 0; j < 4; ++j)
        dst[(size_t)(ty + 8 * j) * Cc + tx] = tile[tx][ty + 8 * j];
}

// ---------------------------------------------------------------------------
// Kernel 3: gather + mean.
//   grid = (P/64, B), block = 256 threads (8 wave32).
//   Phase 0a: wave 0 issues ONE Tensor-Data-Mover op that DMAs this block's
//             64*16 int32 indices (1024-element 1-D tile, 4 KB) into LDS,
//             completed with s_wait_tensorcnt (TENSORcnt path).
//   Phase 0b: belt-and-braces — the same 4 KB is re-copied with the
//             ISA-verified async-to-LDS b128 path (ASYNCcnt), so the LDS
//             index contents are correct even if the D# bit packing (from
//             PDF-extracted tables) is off; the TDM then acts as at least an
//             L2 prefetch. Cost: 4 KB vs 512 KB gather traffic per block.
//   Phase 1:  each wave owns 8 points; per point, lane L accumulates channels
//             4L..4L+3 over 16 votes with wave-coalesced 512 B float4 loads
//             from featsT (L2-resident after first touch).
//   Phase 2:  stage into LDS, write out (B,C,P) coalesced + non-temporal so
//             the 67 MB streaming output does not evict featsT from L2.
// ---------------------------------------------------------------------------
__global__ __launch_bounds__(THREADS)
void ht2s_gather_mean_kernel(const float* __restrict__ featsT,
                             const int* __restrict__ flat_idx,
                             float* __restrict__ out)
{
    __shared__ float sacc[PTS_PER_BLOCK][Cc + 4];          // +4 pad: stride 132
    __shared__ int   sidx[PTS_PER_BLOCK * Vc];             // 4 KB of indices

    const int b  = blockIdx.y;
    const int p0 = blockIdx.x * PTS_PER_BLOCK;
    const int t  = threadIdx.x;

    // ---- Phase 0a: TDM descriptor + tensor_load_to_lds (wave 0 only)
    if (t < 32) {
        const unsigned lds_addr = (unsigned)(size_t)(&sidx[0]);
        const unsigned long long gaddr =
            (unsigned long long)(size_t)(flat_idx + (size_t)p0 * Vc);

        // D# group 0 (128b): count=1 | lds_addr | global_addr[56:0] | type=2
        v4u g0;
        g0.x = 1u;                                   // count=1, user descriptor
        g0.y = lds_addr;                             // LDS byte address
        g0.z = (unsigned)(gaddr & 0xFFFFFFFFu);      // global_addr[31:0]
        g0.w = (unsigned)((gaddr >> 32) & 0x01FFFFFFu) | (2u << 30); // type=2

        // D# group 1 (256b): data_size=4B, tensor_dim0=1024, tensor_dim1=1,
        // tile_dim0=1024, tile_dim1=1, tensor_dim0_stride=1024
        v8i g1;
        g1[0] = (int)(2u << 16);                     // data_size=2 (4 bytes)
        g1[1] = (int)(1024u << 16);                  // tensor_dim0[15:0]
        g1[2] = (int)(1u << 16);                     // dim0 hi=0, tensor_dim1=1
        g1[3] = (int)(1024u << 16);                  // dim1 hi=0, tile_dim0=1024
        g1[4] = 1;                                   // tile_dim1=1, tile_dim2=0
        g1[5] = 1024;                                // tensor_dim0_stride lo
        g1[6] = 0;                                   // stride0 hi, stride1 lo
        g1[7] = 0;                                   // stride1 hi

        v4i gz4 = {0, 0, 0, 0};                      // groups 2/3 unused (<=2D)
        v8i gz8 = {0, 0, 0, 0, 0, 0, 0, 0};         // extra words (6-arg form)
        __builtin_amdgcn_tensor_load_to_lds(g0, g1, gz4, gz4, gz8, 0);
        __builtin_amdgcn_s_wait_tensorcnt(0);
    }
    __syncthreads();   // TDM writes are complete before the verified copy

    // ---- Phase 0b: verified async b128 copy of the same slice (16 B/thread)
    {
        const int* gsrc = flat_idx + (size_t)p0 * Vc + t * 4;
        unsigned lds_off = (unsigned)(size_t)(&sidx[t * 4]);
        asm volatile("global_load_async_to_lds_b128 %0, %1, off"
                     :: "v"(lds_off), "v"(gsrc) : "memory");
        asm volatile("s_wait_asynccnt 0" ::: "memory");
    }
    __syncthreads();

    const int wave = t >> 5;
    const int lane = t & 31;
    const float* base = featsT + (size_t)b * HWc * Cc + lane * 4;

    // ---- Phase 1: 8 points per wave, 16 votes each, float4 per lane
#pragma unroll
    for (int i = 0; i < 8; ++i) {
        const int pl = wave * 8 + i;                 // local point id 0..63
        float ax = 0.f, ay = 0.f, az = 0.f, aw = 0.f;
#pragma unroll
        for (int v = 0; v < Vc; ++v) {
            const int idx = sidx[pl * Vc + v];       // wave-uniform broadcast
            const float4 val =
                *(const float4*)(base + (size_t)idx * Cc);
            ax += val.x; ay += val.y; az += val.z; aw += val.w;
        }
        const float s = 1.0f / 16.0f;
        float4 r; r.x = ax * s; r.y = ay * s; r.z = az * s; r.w = aw * s;
        *(float4*)&sacc[pl][lane * 4] = r;           // 528 B row stride, 16B ok
    }
    __syncthreads();

    // ---- Phase 2: coalesced non-temporal write-out. out[b][c][p], p fastest.
    const size_t outbase = ((size_t)b * Cc) * (size_t)Pc + (size_t)p0;
    const int p = t & 63;
    const int csub = t >> 6;                         // 0..3
#pragma unroll
    for (int cb = 0; cb < 128; cb += 4) {
        const int c = cb + csub;
        __builtin_nontemporal_store(sacc[p][c],
                                    &out[outbase + ((size_t)c << 15) + p]);
    }
}

// ---------------------------------------------------------------------------
extern "C" void kernel_launch(void* const* d_in, const int* in_sizes, int n_in,
                              void* d_out, int out_size, void* d_ws, size_t ws_size,
                              hipStream_t stream)
{
    const float*     feats   = (const float*)d_in[0];
    const long long* mapping = (const long long*)d_in[1];
    float*           out     = (float*)d_out;

    // Workspace layout: [featsT : B*HW*C floats][flat_idx : P*V ints]
    float* featsT  = (float*)d_ws;
    size_t featsT_bytes = (size_t)Bc * HWc * Cc * sizeof(float); // 67.8 MB
    int*   flat_idx = (int*)((char*)d_ws + featsT_bytes);

    const int n_idx = Pc * Vc; // 524288
    ht2s_idx_kernel<<<n_idx / 256, 256, 0, stream>>>(mapping, flat_idx, n_idx);

    ht2s_transpose_kernel<<<dim3(HWc / 32, Cc / 32, Bc), dim3(32, 8), 0, stream>>>(
        feats, featsT);

    ht2s_gather_mean_kernel<<<dim3(Pc / PTS_PER_BLOCK, Bc), THREADS, 0, stream>>>(
        featsT, flat_idx, out);
}